// ProductLayer_31851477467727
// MI455X (gfx1250) — compile-verified
//
#include <hip/hip_runtime.h>
#include <stdint.h>

// Segmented gather-sum:  out[j] = sum_{e: csr[e]==j} x[ptrs[e]],  csr sorted.
// Memory-bound (~350MB HBM traffic -> ~15us floor at 23.3TB/s):
//  - csr/ptrs (256MB) streamed via CDNA5 async global->LDS copies, marked
//    non-temporal so they don't evict the L2-resident x (32MB) that the
//    random gathers depend on.
//  - x gathered 16-deep per lane (regular-temporal: want L2 hits, 4x reuse).
//  - per-segment partials flushed with no-return global_atomic_add_f32.

#define TPB 256                 // threads per block (8 wave32s)
#define EPT 16                  // edges per thread
#define EPB (TPB * EPT)         // 4096 edges per block
#define SENTINEL 0x7FFFFFFF

__global__ void __launch_bounds__(256) zero_kernel(float* __restrict__ out, int n) {
    int i = (blockIdx.x * blockDim.x + threadIdx.x) * 4;
    if (i + 3 < n) {
        *reinterpret_cast<float4*>(out + i) = make_float4(0.f, 0.f, 0.f, 0.f);
    } else {
        for (int k = 0; k < 4; ++k)
            if (i + k < n) out[i + k] = 0.f;
    }
}

// Fire-and-forget FP32 atomic add (no return value -> STOREcnt, no VGPR writeback).
__device__ __forceinline__ void atomic_add_f32_noret(float* p, float v) {
    asm volatile("global_atomic_add_f32 %0, %1, off"
                 :: "v"(p), "v"(v)
                 : "memory");
}

__global__ void __launch_bounds__(TPB) seg_gather_sum_kernel(
    const float* __restrict__ x,
    const int*   __restrict__ ptrs,
    const int*   __restrict__ csr,
    float*       __restrict__ out,
    int n_edges, int n_out)
{
    __shared__ __align__(16) int lds_csr[EPB];   // 16 KB
    __shared__ __align__(16) int lds_ptr[EPB];   // 16 KB

    const int tid = threadIdx.x;
    const long long base = (long long)blockIdx.x * EPB;

    if (base + EPB <= (long long)n_edges) {
        // ---- fast path: CDNA5 async global->LDS bulk copy of the edge tile ----
        // Generic pointer low 32 bits of a __shared__ object == workgroup-relative
        // LDS offset; hardware adds LDS_BASE for async-to-LDS VDST addressing.
        // th:TH_LOAD_NT: the 256MB index stream is touched exactly once -- keep
        // it from thrashing the L2 where x must stay resident.
        const unsigned lcsr = (unsigned)(size_t)lds_csr;
        const unsigned lptr = (unsigned)(size_t)lds_ptr;
        const int* gcsr = csr  + base;   // uniform per block -> SGPR pair
        const int* gptr = ptrs + base;
        const unsigned toff = (unsigned)tid * 16u;  // 16B per lane per issue
#pragma unroll
        for (int k = 0; k < 4; ++k) {
            const unsigned off = toff + (unsigned)k * (TPB * 16u); // 4096B stride
            asm volatile("global_load_async_to_lds_b128 %0, %1, %2 th:TH_LOAD_NT"
                         :: "v"(lcsr + off), "v"(off), "s"(gcsr) : "memory");
            asm volatile("global_load_async_to_lds_b128 %0, %1, %2 th:TH_LOAD_NT"
                         :: "v"(lptr + off), "v"(off), "s"(gptr) : "memory");
        }
        asm volatile("s_wait_asynccnt 0x0" ::: "memory");
        __syncthreads();
    } else {
        // ---- tail path: guarded scalar loads (only the last partial block) ----
        for (int i = tid; i < EPB; i += TPB) {
            long long e = base + i;
            if (e < (long long)n_edges) { lds_csr[i] = csr[e]; lds_ptr[i] = ptrs[e]; }
            else                        { lds_csr[i] = SENTINEL; lds_ptr[i] = 0; }
        }
        __syncthreads();
    }

    // Each thread owns EPT contiguous edges of the (sorted) tile.
    const int s0 = tid * EPT;
    int sg[EPT], pp[EPT];
#pragma unroll
    for (int k = 0; k < EPT / 4; ++k) {
        int4 c = reinterpret_cast<const int4*>(lds_csr + s0)[k];
        int4 p = reinterpret_cast<const int4*>(lds_ptr + s0)[k];
        sg[4*k+0] = c.x; sg[4*k+1] = c.y; sg[4*k+2] = c.z; sg[4*k+3] = c.w;
        pp[4*k+0] = p.x; pp[4*k+1] = p.y; pp[4*k+2] = p.z; pp[4*k+3] = p.w;
    }

    // Issue all 16 random gathers of x up front (16-deep MLP per lane;
    // x is L2-resident at 32MB so most of these hit the 192MB L2).
    float v[EPT];
#pragma unroll
    for (int i = 0; i < EPT; ++i) v[i] = x[pp[i]];

    // Run-length accumulate over sorted segment ids; flush at boundaries.
    int   seg = sg[0];
    float acc = 0.f;
#pragma unroll
    for (int i = 0; i < EPT; ++i) {
        if (sg[i] != seg) {
            if (seg < n_out) atomic_add_f32_noret(out + seg, acc);
            acc = 0.f;
            seg = sg[i];
        }
        acc += v[i];
    }
    if (seg < n_out) atomic_add_f32_noret(out + seg, acc);
}

extern "C" void kernel_launch(void* const* d_in, const int* in_sizes, int n_in,
                              void* d_out, int out_size, void* d_ws, size_t ws_size,
                              hipStream_t stream) {
    const float* x    = (const float*)d_in[0];
    const int*   ptrs = (const int*)d_in[1];
    const int*   csr  = (const int*)d_in[2];
    float*       out  = (float*)d_out;

    const int n_edges = in_sizes[1];
    const int n_out   = out_size;

    // 1) zero the output (harness poisons d_out; empty segments must be 0)
    int zblocks = (n_out + 1023) / 1024;
    zero_kernel<<<zblocks, 256, 0, stream>>>(out, n_out);

    // 2) fused gather + segmented reduction
    int nblocks = (int)(((long long)n_edges + EPB - 1) / EPB);
    seg_gather_sum_kernel<<<nblocks, TPB, 0, stream>>>(x, ptrs, csr, out,
                                                       n_edges, n_out);
}